// InteractionArch_12927851561549
// MI455X (gfx1250) — compile-verified
//
#include <hip/hip_runtime.h>

typedef __attribute__((ext_vector_type(2))) float v2f;
typedef __attribute__((ext_vector_type(8))) float v8f;

#define D_DIM   128
#define NROWS   27          // 1 dense row + 26 sparse rows
#define OUT_COLS 479        // 128 dense + 351 triu interactions
#define SPB     4           // samples per block (1 per wave, 4 waves)
#define ROW_DW  132         // 128 + 4 dword pad -> conflict-free 64b LDS reads
#define ROWS_AL 28          // allocated rows per sample slab

// triu flat index over 27x27, k=1: pairs (f,g), f<g
__device__ __forceinline__ int triu_idx(int f, int g) {
    return f * 26 - (f * (f - 1)) / 2 + (g - f - 1);
}

__global__ __launch_bounds__(128)
void dlrm_interact_kernel(const float* __restrict__ dense,
                          const float* __restrict__ sparse,
                          float* __restrict__ out)
{
    __shared__ __align__(16) float smem[SPB * ROWS_AL * ROW_DW];

    const int tid      = threadIdx.x;
    const int block_b0 = blockIdx.x * SPB;

    // ---- cooperative staging: 4 samples x 27 rows x 128 floats, float4 coalesced.
    // 4*27*32 = 3456 float4 chunks over 128 threads -> 27 iterations.
    #pragma unroll
    for (int it = 0; it < 27; ++it) {
        int t   = tid + it * 128;        // 0..3455
        int s   = t / (NROWS * 32);
        int rem = t % (NROWS * 32);
        int row = rem / 32;
        int cv  = rem % 32;              // float4 column index
        int b   = block_b0 + s;
        const float4* src;
        if (row == 0)
            src = (const float4*)(dense + (size_t)b * D_DIM) + cv;
        else
            src = (const float4*)(sparse + ((size_t)b * 26 + (row - 1)) * D_DIM) + cv;
        float4 v = *src;
        *(float4*)&smem[(s * ROWS_AL + row) * ROW_DW + cv * 4] = v; // 528B row stride keeps 16B align
    }

    // ---- dense passthrough into out[:, 0:128] (coalesced scalar stores; row
    // stride 479 floats is odd so vector stores would be misaligned anyway)
    #pragma unroll
    for (int s = 0; s < SPB; ++s) {
        int b = block_b0 + s;
        out[(size_t)b * OUT_COLS + tid] = dense[(size_t)b * D_DIM + tid];
    }

    __syncthreads();

    // ---- per-wave Gram matrix via V_WMMA_F32_16X16X4_F32
    const int wave = tid >> 5;           // sample within block
    const int lane = tid & 31;
    const int b    = block_b0 + wave;
    const int mrow = lane & 15;          // M (and N) index of this lane's fragment
    const int koff = (lane >> 4) * 2;    // K sub-offset: lanes 16-31 hold K+2,K+3
    const float* base = &smem[wave * ROWS_AL * ROW_DW];

    v8f acc00 = {};  // rows 0-15  x rows 0-15
    v8f acc01 = {};  // rows 0-15  x rows 16-26
    v8f acc11 = {};  // rows 16-26 x rows 16-26

    #pragma unroll 4
    for (int kk = 0; kk < 32; ++kk) {
        int col = kk * 4 + koff;
        // A(16x4) and B(4x16) f32 fragments share the same lane layout, so one
        // row-block load serves as both operands of X * X^T.
        v2f a0 = *(const v2f*)(base +  mrow       * ROW_DW + col);
        v2f a1 = *(const v2f*)(base + (mrow + 16) * ROW_DW + col); // rows >=27: garbage, never stored
        acc00 = __builtin_amdgcn_wmma_f32_16x16x4_f32(false, a0, false, a0, (short)0, acc00, false, false);
        acc01 = __builtin_amdgcn_wmma_f32_16x16x4_f32(false, a0, false, a1, (short)0, acc01, false, false);
        acc11 = __builtin_amdgcn_wmma_f32_16x16x4_f32(false, a1, false, a1, (short)0, acc11, false, false);
    }

    // ---- scatter upper triangle. Accumulator layout: lane L holds N = L&15,
    // VGPR v holds M = v + 8*(L>>4). Tile (I,J): G[f=16I+M, g=16J+N].
    float* orow = out + (size_t)b * OUT_COLS + D_DIM;
    const int N = lane & 15;
    const int mhi = (lane >> 4) << 3;
    #pragma unroll
    for (int v = 0; v < 8; ++v) {
        int M = v + mhi;
        if (M < N)                 orow[triu_idx(M,      N     )] = acc00[v];
        if (16 + N <= 26)          orow[triu_idx(M,      16 + N)] = acc01[v];
        if (M < N && 16 + N <= 26) orow[triu_idx(16 + M, 16 + N)] = acc11[v];
    }
}

extern "C" void kernel_launch(void* const* d_in, const int* in_sizes, int n_in,
                              void* d_out, int out_size, void* d_ws, size_t ws_size,
                              hipStream_t stream) {
    const float* dense  = (const float*)d_in[0];   // [B, 128]
    const float* sparse = (const float*)d_in[1];   // [B, 26, 128]
    float* out = (float*)d_out;                    // [B, 479]
    int B = in_sizes[0] / D_DIM;                   // 32768 (multiple of SPB)
    int grid = B / SPB;
    dlrm_interact_kernel<<<grid, 128, 0, stream>>>(dense, sparse, out);
}